// ModuleDSepconvCPU_44547400794794
// MI455X (gfx1250) — compile-verified
//
#include <hip/hip_runtime.h>
#include <hip/hip_bf16.h>
#include <stdint.h>

// Problem dims (fixed by the reference)
#define BB 4
#define CC 3
#define FF 5
#define HOUT 256
#define WOUT 256
#define HIN 260
#define WIN 260
#define CH (HIN * WIN)          // 67600, per-channel input stride
#define NTAP (FF * FF)          // 25
#define BLK 256

// ---- CDNA5 async global->LDS copy (per-lane 4B), tracked by ASYNCcnt ----
// Non-temporal: streamed data is consumed exactly once.
__device__ __forceinline__ void async_ld_b32(uint32_t lds_byte_addr, const float* gaddr) {
    asm volatile("global_load_async_to_lds_b32 %0, %1, off th:TH_LOAD_NT"
                 :
                 : "v"(lds_byte_addr), "v"(gaddr)
                 : "memory");
}
__device__ __forceinline__ void wait_async_le6() {
    asm volatile("s_wait_asynccnt 0x6" ::: "memory");
}
__device__ __forceinline__ void wait_async_le3() {
    asm volatile("s_wait_asynccnt 0x3" ::: "memory");
}
__device__ __forceinline__ void wait_async_le0() {
    asm volatile("s_wait_asynccnt 0x0" ::: "memory");
}

__global__ __launch_bounds__(BLK) void dsepconv_kernel(
    const float* __restrict__ inp,   // (B,C,HIN,WIN)
    const float* __restrict__ vert,  // (B,F,HO,WO)
    const float* __restrict__ horz,  // (B,F,HO,WO)
    const float* __restrict__ offx,  // (B,F*F,HO,WO)  -> drives py
    const float* __restrict__ offy,  // (B,F*F,HO,WO)  -> drives px
    const float* __restrict__ mask,  // (B,F*F,HO,WO)
    float* __restrict__ out)         // (B,C,HO,WO)
{
    // Triple-buffered LDS staging (depth-2 async prefetch pipeline).
    __shared__ float sOX[3][BLK];
    __shared__ float sOY[3][BLK];
    __shared__ float sM [3][BLK];

    const int tid = threadIdx.x;
    const int pix = blockIdx.x * BLK + tid;     // b*65536 + h*256 + w (exact fit)
    const int b   = pix >> 16;
    const int h   = (pix >> 8) & 255;
    const int w   = pix & 255;
    const int hw  = (h << 8) | w;

    const float* vb  = vert + ((b * FF)   << 16) + hw;
    const float* hb  = horz + ((b * FF)   << 16) + hw;
    const float* oxb = offx + ((b * NTAP) << 16) + hw;
    const float* oyb = offy + ((b * NTAP) << 16) + hw;
    const float* mb  = mask + ((b * NTAP) << 16) + hw;
    const float* ib  = inp + (size_t)b * (CC * CH);

    // Warm L2 with the nominal gather neighborhood (gfx1250 global_prefetch_b8).
#pragma unroll
    for (int c = 0; c < CC; ++c)
        __builtin_prefetch(ib + c * CH + h * WIN + w, 0, 0);

    // Separable weights stay in registers (fully-unrolled tap loop keeps
    // v[fy]/hz[fx] statically indexed).
    float v[FF], hz[FF];
#pragma unroll
    for (int f = 0; f < FF; ++f) {
        v[f]  = vb[f << 16];
        hz[f] = hb[f << 16];
    }

    // LDS byte addresses of this thread's private slots. Lane-owned
    // producer/consumer: lane i of a wave async-copies exactly the element
    // thread i of that wave later reads -> per-wave ASYNCcnt suffices,
    // no workgroup barrier needed.
    uint32_t aOX[3], aOY[3], aM[3];
#pragma unroll
    for (int k = 0; k < 3; ++k) {
        aOX[k] = (uint32_t)(uintptr_t)&sOX[k][tid];
        aOY[k] = (uint32_t)(uintptr_t)&sOY[k][tid];
        aM [k] = (uint32_t)(uintptr_t)&sM [k][tid];
    }

    // Stage taps 0 and 1 (pipeline depth 2).
    async_ld_b32(aOX[0], oxb);
    async_ld_b32(aOY[0], oyb);
    async_ld_b32(aM [0], mb);
    async_ld_b32(aOX[1], oxb + (1 << 16));
    async_ld_b32(aOY[1], oyb + (1 << 16));
    async_ld_b32(aM [1], mb  + (1 << 16));

    float acc0 = 0.0f, acc1 = 0.0f, acc2 = 0.0f;

#pragma unroll
    for (int tap = 0; tap < NTAP; ++tap) {
        const int buf = tap % 3;
        if (tap + 2 < NTAP) {
            // Stage tap+2 into the buffer freed at iteration tap-1, then wait
            // until only the newest 6 (taps tap+1, tap+2) are outstanding.
            // Async loads complete in order -> tap's 3 values are in LDS.
            const int nb = (tap + 2) % 3;
            async_ld_b32(aOX[nb], oxb + ((tap + 2) << 16));
            async_ld_b32(aOY[nb], oyb + ((tap + 2) << 16));
            async_ld_b32(aM [nb], mb  + ((tap + 2) << 16));
            wait_async_le6();
        } else if (tap + 1 < NTAP) {
            wait_async_le3();
        } else {
            wait_async_le0();
        }

        const float ox = sOX[buf][tid];   // -> py
        const float oy = sOY[buf][tid];   // -> px
        const float m  = sM [buf][tid];

        const int fy = tap / FF;
        const int fx = tap - fy * FF;

        // px = clip(offY + w + fx - half + 1, 0, WIN-1), half = 2
        float px = oy + (float)(w + fx - 1);
        float py = ox + (float)(h + fy - 1);
        px = fminf(fmaxf(px, 0.0f), (float)(WIN - 1));
        py = fminf(fmaxf(py, 0.0f), (float)(HIN - 1));

        const float lf = floorf(px);
        const float tf = floorf(py);
        const float wx = 1.0f - (px - lf);
        const float wy = 1.0f - (py - tf);

        const int l = (int)lf;            // in [0, WIN-1]
        const int t = (int)tf;            // in [0, HIN-1]

        // Border-aware corner addressing with conditional increments:
        // r = min(l+1, WIN-1) -> +inc_r ; bt = min(t+1, HIN-1) -> +inc_b.
        const int inc_r = (l < WIN - 1) ? 1 : 0;
        const int inc_b = (t < HIN - 1) ? WIN : 0;
        const int a_tl  = t * WIN + l;
        const int a_tr  = a_tl + inc_r;
        const int a_bl  = a_tl + inc_b;
        const int a_br  = a_bl + inc_r;

        const float wtap = v[fy] * hz[fx] * m;
        const float w_tl = wx * wy * wtap;
        const float w_tr = (1.0f - wx) * wy * wtap;
        const float w_bl = wx * (1.0f - wy) * wtap;
        const float w_br = (1.0f - wx) * (1.0f - wy) * wtap;

        // Channel 0
        acc0 = fmaf(ib[a_tl],          w_tl, acc0);
        acc0 = fmaf(ib[a_tr],          w_tr, acc0);
        acc0 = fmaf(ib[a_bl],          w_bl, acc0);
        acc0 = fmaf(ib[a_br],          w_br, acc0);
        // Channel 1 (immediate offset CH*4 fits signed-24-bit IOFFSET)
        acc1 = fmaf(ib[a_tl + CH],     w_tl, acc1);
        acc1 = fmaf(ib[a_tr + CH],     w_tr, acc1);
        acc1 = fmaf(ib[a_bl + CH],     w_bl, acc1);
        acc1 = fmaf(ib[a_br + CH],     w_br, acc1);
        // Channel 2
        acc2 = fmaf(ib[a_tl + 2 * CH], w_tl, acc2);
        acc2 = fmaf(ib[a_tr + 2 * CH], w_tr, acc2);
        acc2 = fmaf(ib[a_bl + 2 * CH], w_bl, acc2);
        acc2 = fmaf(ib[a_br + 2 * CH], w_br, acc2);
    }

    out[((b * CC + 0) << 16) + hw] = acc0;
    out[((b * CC + 1) << 16) + hw] = acc1;
    out[((b * CC + 2) << 16) + hw] = acc2;
}

extern "C" void kernel_launch(void* const* d_in, const int* in_sizes, int n_in,
                              void* d_out, int out_size, void* d_ws, size_t ws_size,
                              hipStream_t stream) {
    const float* inp  = (const float*)d_in[0];
    const float* vert = (const float*)d_in[1];
    const float* horz = (const float*)d_in[2];
    const float* offx = (const float*)d_in[3];
    const float* offy = (const float*)d_in[4];
    const float* mask = (const float*)d_in[5];
    float* out = (float*)d_out;

    const int npix = BB * HOUT * WOUT;          // 262144, exact multiple of BLK
    dsepconv_kernel<<<dim3(npix / BLK), dim3(BLK), 0, stream>>>(
        inp, vert, horz, offx, offy, mask, out);
}